// GraphFilm_68453188763818
// MI455X (gfx1250) — compile-verified
//
#include <hip/hip_runtime.h>

#define NNODES   100000
#define NEDGES   1600000
#define NGRAPHS  64
#define NODE_DIM 256
#define COND_DIM 512
#define OUT_DIM  128
#define HID      64
#define EPS      1e-5f

typedef __attribute__((ext_vector_type(2))) float v2f;
typedef __attribute__((ext_vector_type(8))) float v8f;

__device__ __forceinline__ v8f wmma4(v2f a, v2f b, v8f c) {
    // V_WMMA_F32_16X16X4_F32 : D[16x16] = A[16x4] * B[4x16] + C
    return __builtin_amdgcn_wmma_f32_16x16x4_f32(
        /*neg_a=*/false, a, /*neg_b=*/false, b,
        /*c_mod=*/(short)0, c, /*reuse_a=*/false, /*reuse_b=*/false);
}

// ---------------------------------------------------------------------------
// cond projection: cp = cond_feats @ wn(cond_w_v, cond_w_g)^T + cond_b
// gb[b*128 + o] = cp[b][o] + (o < 64 ? 1 : 0)   (gamma pre-offset applied)
// ---------------------------------------------------------------------------
__global__ void cond_kernel(const float* __restrict__ cf,
                            const float* __restrict__ cwv,
                            const float* __restrict__ cwg,
                            const float* __restrict__ cb,
                            float* __restrict__ gb) {
    const int o = blockIdx.x;              // 0..127
    const int t = threadIdx.x;             // 128 threads
    __shared__ float row[COND_DIM];
    __shared__ float red[4];

    float ss = 0.0f;
    for (int i = t; i < COND_DIM; i += 128) {
        float v = cwv[o * COND_DIM + i];
        row[i] = v;
        ss += v * v;
    }
    #pragma unroll
    for (int m = 1; m < 32; m <<= 1) ss += __shfl_xor(ss, m, 32);
    if ((t & 31) == 0) red[t >> 5] = ss;
    __syncthreads();
    const float scale = cwg[o] / sqrtf(red[0] + red[1] + red[2] + red[3]);

    if (t < NGRAPHS) {
        const float* c = cf + t * COND_DIM;
        float dot = 0.0f;
        #pragma unroll 8
        for (int k = 0; k < COND_DIM; ++k) dot += c[k] * row[k];
        float val = dot * scale + cb[o];
        gb[t * OUT_DIM + o] = val + (o < HID ? 1.0f : 0.0f);
    }
}

// ---------------------------------------------------------------------------
// wl = wn(lin_w_v, lin_w_g)   [128 x 64]
// ---------------------------------------------------------------------------
__global__ void linwn_kernel(const float* __restrict__ wv,
                             const float* __restrict__ wg,
                             float* __restrict__ wl) {
    const int o = blockIdx.x;   // 0..127
    const int t = threadIdx.x;  // 32
    float a = wv[o * HID + t];
    float b = wv[o * HID + 32 + t];
    float ss = a * a + b * b;
    #pragma unroll
    for (int m = 1; m < 32; m <<= 1) ss += __shfl_xor(ss, m, 32);
    const float scale = wg[o] / sqrtf(ss);
    wl[o * HID + t]      = a * scale;
    wl[o * HID + 32 + t] = b * scale;
}

// ---------------------------------------------------------------------------
// Main fused node pipeline: GEMM1(256->64) -> LN -> FiLM -> ReLU -> GEMM2(64->128)
// One wave per 16-node tile; WMMA f32 16x16x4 throughout.
// ---------------------------------------------------------------------------
__global__ __launch_bounds__(256) void film_main(
        const float* __restrict__ nf, const float* __restrict__ fw,
        const float* __restrict__ fb, const float* __restrict__ wl,
        const float* __restrict__ lb, const float* __restrict__ gb,
        const int*   __restrict__ n2g, float* __restrict__ x2) {
    __shared__ float lds[8 * 16 * HID];          // 8 waves * 16x64 tile = 32 KB
    const int lane = threadIdx.x & 31;
    const int wave = threadIdx.x >> 5;
    const int tile = blockIdx.x * 8 + wave;
    if (tile * 16 >= NNODES) return;             // wave-uniform; EXEC stays all-ones
    const int node0 = tile * 16;
    const int half  = lane >> 4;                 // 0: K even pair, 1: K odd pair
    const int l16   = lane & 15;
    const int koff  = 2 * half;

    // ---- GEMM1: h[16x64] = nf[16x256] @ fw^T, 4 N-tiles, K in steps of 4 ----
    v8f acc[4] = {};
    const float* arow = nf + (node0 + l16) * NODE_DIM + koff;
    #pragma unroll 4
    for (int k = 0; k < NODE_DIM; k += 4) {
        v2f a = *(const v2f*)(arow + k);
        #pragma unroll
        for (int nt = 0; nt < 4; ++nt) {
            v2f b = *(const v2f*)(fw + (nt * 16 + l16) * NODE_DIM + k + koff);
            acc[nt] = wmma4(a, b, acc[nt]);
        }
    }

    // ---- bias + LayerNorm(64) + FiLM + ReLU; element (r,nt): row r+8*half, col nt*16+l16
    float* hl = lds + wave * (16 * HID);
    #pragma unroll
    for (int r = 0; r < 8; ++r) {
        float v0 = acc[0][r] + fb[ 0 + l16];
        float v1 = acc[1][r] + fb[16 + l16];
        float v2 = acc[2][r] + fb[32 + l16];
        float v3 = acc[3][r] + fb[48 + l16];
        float s = v0 + v1 + v2 + v3;
        float q = v0 * v0 + v1 * v1 + v2 * v2 + v3 * v3;
        #pragma unroll
        for (int m = 1; m < 16; m <<= 1) {       // reduce within each 16-lane half
            s += __shfl_xor(s, m, 32);
            q += __shfl_xor(q, m, 32);
        }
        const float mu   = s * (1.0f / HID);
        const float var  = q * (1.0f / HID) - mu * mu;
        const float rstd = rsqrtf(var + EPS);
        const int   node = node0 + r + 8 * half;
        const int   g    = n2g[node];
        const float* gbase = gb + g * OUT_DIM;   // [0:64)=gamma(+1), [64:128)=beta
        float h0 = fmaxf(gbase[ 0 + l16] * ((v0 - mu) * rstd) + gbase[64 +  0 + l16], 0.0f);
        float h1 = fmaxf(gbase[16 + l16] * ((v1 - mu) * rstd) + gbase[64 + 16 + l16], 0.0f);
        float h2 = fmaxf(gbase[32 + l16] * ((v2 - mu) * rstd) + gbase[64 + 32 + l16], 0.0f);
        float h3 = fmaxf(gbase[48 + l16] * ((v3 - mu) * rstd) + gbase[64 + 48 + l16], 0.0f);
        float* hr = hl + (r + 8 * half) * HID;
        hr[ 0 + l16] = h0; hr[16 + l16] = h1; hr[32 + l16] = h2; hr[48 + l16] = h3;
    }
    // per-wave LDS transpose handoff (same-wave DS ops are in-order; wait to be safe)
    asm volatile("s_wait_dscnt 0" ::: "memory");

    // ---- GEMM2: x2[16x128] = h[16x64] @ wl^T, 8 N-tiles ----
    v8f acc2[8] = {};
    #pragma unroll 4
    for (int k = 0; k < HID; k += 4) {
        v2f a = *(const v2f*)(hl + l16 * HID + k + koff);
        #pragma unroll
        for (int nt = 0; nt < 8; ++nt) {
            v2f b = *(const v2f*)(wl + (nt * 16 + l16) * HID + k + koff);
            acc2[nt] = wmma4(a, b, acc2[nt]);
        }
    }
    #pragma unroll
    for (int nt = 0; nt < 8; ++nt) {
        const float bias = lb[nt * 16 + l16];
        #pragma unroll
        for (int r = 0; r < 8; ++r) {
            const int node = node0 + r + 8 * half;
            x2[node * OUT_DIM + nt * 16 + l16] = acc2[nt][r] + bias;
        }
    }
}

// ---------------------------------------------------------------------------
// zero accumulators (d_out / cnt are poisoned before timing)
// ---------------------------------------------------------------------------
__global__ void zero_kernel(float* __restrict__ out, float* __restrict__ cnt) {
    const int t = blockIdx.x * blockDim.x + threadIdx.x;
    if (t < NNODES * OUT_DIM) out[t] = 0.0f;
    if (t < NNODES) cnt[t] = 0.0f;
}

// ---------------------------------------------------------------------------
// edge aggregation: out[dst] += x2[src]; cnt[dst] += 1   (L2-resident atomics)
// ---------------------------------------------------------------------------
__global__ void edge_kernel(const float* __restrict__ x2, const int* __restrict__ ei,
                            float* __restrict__ out, float* __restrict__ cnt) {
    const long long t = (long long)blockIdx.x * blockDim.x + threadIdx.x;
    const int e = (int)(t >> 5);
    if (e >= NEDGES) return;
    const int lane = (int)t & 31;
    const int src = ei[e];
    const int dst = ei[NEDGES + e];
    const float4 v = ((const float4*)(x2 + (long long)src * OUT_DIM))[lane];
    float* o = out + (long long)dst * OUT_DIM + lane * 4;
    atomicAdd(o + 0, v.x);
    atomicAdd(o + 1, v.y);
    atomicAdd(o + 2, v.z);
    atomicAdd(o + 3, v.w);
    if (lane == 0) atomicAdd(cnt + dst, 1.0f);
}

// ---------------------------------------------------------------------------
// finalize: mean + relu (in place on d_out)
// ---------------------------------------------------------------------------
__global__ void finalize_kernel(float* __restrict__ out, const float* __restrict__ cnt) {
    const int t = blockIdx.x * blockDim.x + threadIdx.x;
    if (t >= NNODES * OUT_DIM) return;
    const float c = fmaxf(cnt[t >> 7], 1.0f);
    out[t] = fmaxf(out[t] / c, 0.0f);
}

extern "C" void kernel_launch(void* const* d_in, const int* in_sizes, int n_in,
                              void* d_out, int out_size, void* d_ws, size_t ws_size,
                              hipStream_t stream) {
    const float* nf  = (const float*)d_in[0];   // node_feats   [100000,256]
    const float* cf  = (const float*)d_in[1];   // cond_feats   [64,512]
    const float* cwv = (const float*)d_in[2];   // cond_w_v     [128,512]
    const float* cwg = (const float*)d_in[3];   // cond_w_g     [128]
    const float* cb  = (const float*)d_in[4];   // cond_b       [128]
    const float* fw  = (const float*)d_in[5];   // film_w       [64,256]
    const float* fb  = (const float*)d_in[6];   // film_b       [64]
    const float* lwv = (const float*)d_in[7];   // lin_w_v      [128,64]
    const float* lwg = (const float*)d_in[8];   // lin_w_g      [128]
    const float* lb  = (const float*)d_in[9];   // lin_b        [128]
    const int*   ei  = (const int*)d_in[10];    // edge_index   [2,1600000]
    const int*   n2g = (const int*)d_in[11];    // node2graph   [100000]
    float* out = (float*)d_out;

    float* ws  = (float*)d_ws;
    float* gb  = ws;                                  // 64*128
    float* wl  = gb + NGRAPHS * OUT_DIM;              // 128*64
    float* cnt = wl + OUT_DIM * HID;                  // 100000
    float* x2  = cnt + ((NNODES + 127) & ~127);       // 100000*128

    cond_kernel<<<OUT_DIM, 128, 0, stream>>>(cf, cwv, cwg, cb, gb);
    linwn_kernel<<<OUT_DIM, 32, 0, stream>>>(lwv, lwg, wl);
    zero_kernel<<<(NNODES * OUT_DIM + 255) / 256, 256, 0, stream>>>(out, cnt);

    const int ntiles = NNODES / 16;                   // 6250, exact
    film_main<<<(ntiles + 7) / 8, 256, 0, stream>>>(nf, fw, fb, wl, lb, gb, n2g, x2);

    const long long ethreads = (long long)NEDGES * 32;
    edge_kernel<<<(int)((ethreads + 255) / 256), 256, 0, stream>>>(x2, ei, out, cnt);
    finalize_kernel<<<(NNODES * OUT_DIM + 255) / 256, 256, 0, stream>>>(out, cnt);
}